// ClasswiseECELoss_32856499815085
// MI455X (gfx1250) — compile-verified
//
#include <hip/hip_runtime.h>
#include <hip/hip_bf16.h>
#include <cstdint>

// Classwise ECE: N=1048576 rows, C=64 classes, 15 bins.
// contrib(seg) = |sum_conf - sum_corr| / N when count>0 (count cancels),
// and count>0 <=> sum_conf>0, so only two 960-entry histograms are needed.
// Memory-bound: 256MB logits streamed once (~11us @ 23.3TB/s). CDNA5 path:
// async global->LDS double-buffering (ASYNCcnt) + wave32 LDS f32 atomics.

#define NBINS 15
#define NCLS  64
#define NSEG  (NCLS * NBINS)   // 960
#define WAVES 8                // 256-thread block = 8 wave32
#define TILE  8                // rows per async tile: 8*64*4B = 2048B contiguous

// ---- CDNA5 async global->LDS copy (tracked by ASYNCcnt) ----
// global_load_async_to_lds_b128 vDstLdsOff, vAddr64, off offset:IMM th:TH_LOAD_NT
// Same IMM offset applies to both LDS and global addresses (ISA 08 §4.4).
// NT hint: 256MB single-pass stream > 192MB L2, zero reuse -> don't thrash L2.
#define ASYNC_CP16(ldsoff, gaddr, OFF)                                         \
  asm volatile("global_load_async_to_lds_b128 %0, %1, off offset:" #OFF        \
               " th:TH_LOAD_NT"                                                \
               :: "v"(ldsoff), "v"(gaddr) : "memory")

#define WAIT_ASYNC_0() asm volatile("s_wait_asynccnt 0x0" ::: "memory")
#define WAIT_ASYNC_4() asm volatile("s_wait_asynccnt 0x4" ::: "memory")

// Low 32 bits of a generic pointer to __shared__ data == LDS byte offset
// (local->flat addrspacecast places the LDS offset in the low dword).
__device__ __forceinline__ uint32_t lds_off(const void* p) {
  return (uint32_t)(uintptr_t)p;
}

__global__ void ece_zero_ws(float* ws, int n) {
  int i = blockIdx.x * blockDim.x + threadIdx.x;
  if (i < n) ws[i] = 0.0f;
}

__global__ __launch_bounds__(256) void ece_hist_kernel(
    const float* __restrict__ logits, const int* __restrict__ labels,
    float* __restrict__ g_conf, float* __restrict__ g_corr, int nRows) {
  __shared__ float h_conf[NSEG];
  __shared__ float h_corr[NSEG];
  __shared__ __align__(16) float stage[WAVES][2][TILE * NCLS];  // 32 KB

  const int tid = threadIdx.x;
  for (int i = tid; i < NSEG; i += blockDim.x) { h_conf[i] = 0.0f; h_corr[i] = 0.0f; }
  __syncthreads();

  const int lane   = tid & 31;
  const int wv     = tid >> 5;
  const int waveId = blockIdx.x * WAVES + wv;
  const int nWaves = gridDim.x * WAVES;
  const int nTiles = nRows / TILE;  // N is a multiple of TILE (2^20 rows)

  int buf  = 0;
  int tile = waveId;

  if (tile < nTiles) {  // prologue: prefetch first tile (4 x 16B x 32 lanes = 2048B)
    const char* g = (const char*)(logits + (size_t)tile * TILE * NCLS) + lane * 16;
    uint32_t    l = lds_off(&stage[wv][0][0]) + (uint32_t)(lane * 16);
    ASYNC_CP16(l, g, 0); ASYNC_CP16(l, g, 512); ASYNC_CP16(l, g, 1024); ASYNC_CP16(l, g, 1536);
  }

  for (; tile < nTiles; tile += nWaves) {
    const int next = tile + nWaves;
    if (next < nTiles) {  // prefetch next tile into the other buffer
      const char* g = (const char*)(logits + (size_t)next * TILE * NCLS) + lane * 16;
      uint32_t    l = lds_off(&stage[wv][buf ^ 1][0]) + (uint32_t)(lane * 16);
      ASYNC_CP16(l, g, 0); ASYNC_CP16(l, g, 512); ASYNC_CP16(l, g, 1024); ASYNC_CP16(l, g, 1536);
      WAIT_ASYNC_4();     // current tile's 4 ops (in-order on ASYNCcnt) are done
    } else {
      WAIT_ASYNC_0();
    }

    const int row0 = tile * TILE;
    int labv = 0;
    if (lane < TILE) labv = labels[row0 + lane];

    const float2* sp = (const float2*)&stage[wv][buf][0];
    const int c0 = lane * 2, c1 = c0 + 1;
#pragma unroll
    for (int r = 0; r < TILE; ++r) {
      float2 x = sp[r * 32 + lane];           // ds_load_b64, conflict-free
      float e0 = __expf(x.x), e1 = __expf(x.y);
      float s = e0 + e1;
#pragma unroll
      for (int o = 16; o > 0; o >>= 1) s += __shfl_xor(s, o, 32);  // wave32 sum of 64 exps
      float inv = 1.0f / s;
      float p0 = e0 * inv, p1 = e1 * inv;
      int b0 = (int)ceilf(p0 * 15.0f) - 1; b0 = b0 < 0 ? 0 : (b0 > 14 ? 14 : b0);
      int b1 = (int)ceilf(p1 * 15.0f) - 1; b1 = b1 < 0 ? 0 : (b1 > 14 ? 14 : b1);
      atomicAdd(&h_conf[c0 * NBINS + b0], p0);  // seg%64 distinct per lane: gcd(15,64)=1
      atomicAdd(&h_conf[c1 * NBINS + b1], p1);
      int lab = __shfl(labv, r, 32);
      if (lab == c0)      atomicAdd(&h_corr[c0 * NBINS + b0], 1.0f);
      else if (lab == c1) atomicAdd(&h_corr[c1 * NBINS + b1], 1.0f);
    }
    buf ^= 1;
  }

  __syncthreads();
  for (int i = tid; i < NSEG; i += blockDim.x) {
    float cf = h_conf[i], cr = h_corr[i];
    if (cf != 0.0f) atomicAdd(&g_conf[i], cf);
    if (cr != 0.0f) atomicAdd(&g_corr[i], cr);
  }
}

__global__ void ece_finalize(const float* __restrict__ g_conf,
                             const float* __restrict__ g_corr,
                             float* __restrict__ out, float invNC) {
  __shared__ float red[256];
  const int tid = threadIdx.x;
  float s = 0.0f;
  for (int i = tid; i < NSEG; i += 256) {
    float cf = g_conf[i];
    if (cf > 0.0f) s += fabsf(cf - g_corr[i]);  // count>0 <=> conf>0
  }
  red[tid] = s;
  __syncthreads();
  for (int o = 128; o > 0; o >>= 1) {
    if (tid < o) red[tid] += red[tid + o];
    __syncthreads();
  }
  if (tid == 0) out[0] = red[0] * invNC;
}

extern "C" void kernel_launch(void* const* d_in, const int* in_sizes, int n_in,
                              void* d_out, int out_size, void* d_ws, size_t ws_size,
                              hipStream_t stream) {
  const float* logits = (const float*)d_in[0];
  const int*   labels = (const int*)d_in[1];
  const int N = in_sizes[1];           // labels: one per row
  float* g_conf = (float*)d_ws;        // 960 floats
  float* g_corr = g_conf + NSEG;       // 960 floats (7680B total scratch)

  ece_zero_ws<<<(2 * NSEG + 255) / 256, 256, 0, stream>>>(g_conf, 2 * NSEG);
  // 1024 blocks x 8 wave32 = 8192 waves; 131072 tiles -> 16 tiles/wave.
  ece_hist_kernel<<<1024, 256, 0, stream>>>(logits, labels, g_conf, g_corr, N);
  ece_finalize<<<1, 256, 0, stream>>>(g_conf, g_corr, (float*)d_out,
                                      1.0f / ((float)N * (float)NCLS));
}